// Layer_71863392797070
// MI455X (gfx1250) — compile-verified
//
#include <hip/hip_runtime.h>
#include <hip/hip_bf16.h>

#define NTOT 65536
#define MDIM 1024
#define DOUT 8
#define NT 32          // columns per workgroup in gp_main
#define KPAD 1028      // padded LDS row stride (floats) for kzx^T
#define JITTER 0.001f

typedef __attribute__((ext_vector_type(2))) float v2f;
typedef __attribute__((ext_vector_type(8))) float v8f;

// ---------------------------------------------------------------- build Kzz
__global__ __launch_bounds__(256) void build_kzz(const float* __restrict__ z,
                                                 const float* __restrict__ ls,
                                                 float* __restrict__ K) {
  int i = blockIdx.x;
  float zi = z[i];
  float inv = 1.0f / ls[0];
  for (int j = threadIdx.x; j < MDIM; j += 256) {
    float d = (zi - z[j]) * inv;
    float v = expf(-0.5f * d * d);
    if (j == i) v += JITTER;
    K[(size_t)i * MDIM + j] = v;
  }
}

// ------------------------------------------------- blocked Cholesky, B = 128
__global__ __launch_bounds__(256) void chol_diag(float* __restrict__ L, int jb) {
  extern __shared__ float lds[];
  float* s = lds;  // [128][129]
  int tid = threadIdx.x;
  int b0 = jb * 128;
  for (int i = tid; i < 128 * 128; i += 256) {
    int r = i >> 7, c = i & 127;
    s[r * 129 + c] = L[(size_t)(b0 + r) * MDIM + b0 + c];
  }
  __syncthreads();
  for (int k = 0; k < 128; ++k) {
    if (tid == 0) s[k * 129 + k] = sqrtf(s[k * 129 + k]);
    __syncthreads();
    float dk = s[k * 129 + k];
    for (int i = k + 1 + tid; i < 128; i += 256) s[i * 129 + k] /= dk;
    __syncthreads();
    for (int j = k + 1; j < 128; ++j) {
      float ljk = s[j * 129 + k];
      for (int i = j + tid; i < 128; i += 256)
        s[i * 129 + j] -= s[i * 129 + k] * ljk;
    }
    __syncthreads();
  }
  for (int i = tid; i < 128 * 128; i += 256) {
    int r = i >> 7, c = i & 127;
    L[(size_t)(b0 + r) * MDIM + b0 + c] = s[r * 129 + c];
  }
}

__global__ __launch_bounds__(128) void chol_trsm(float* __restrict__ L, int jb) {
  extern __shared__ float lds[];
  float* s11 = lds;              // [128][129]
  float* X = lds + 128 * 129;    // [128][129]
  int tid = threadIdx.x;
  int b0 = jb * 128;
  int r0 = (jb + 1 + blockIdx.x) * 128;
  for (int i = tid; i < 128 * 128; i += 128) {
    int r = i >> 7, c = i & 127;
    s11[r * 129 + c] = L[(size_t)(b0 + r) * MDIM + b0 + c];
    X[r * 129 + c]   = L[(size_t)(r0 + r) * MDIM + b0 + c];
  }
  __syncthreads();
  int r = tid;  // one row per thread
  for (int k = 0; k < 128; ++k) {
    float acc = X[r * 129 + k];
    for (int p = 0; p < k; ++p) acc -= X[r * 129 + p] * s11[k * 129 + p];
    X[r * 129 + k] = acc / s11[k * 129 + k];
  }
  __syncthreads();
  for (int i = tid; i < 128 * 128; i += 128) {
    int r2 = i >> 7, c = i & 127;
    L[(size_t)(r0 + r2) * MDIM + b0 + c] = X[r2 * 129 + c];
  }
}

__global__ __launch_bounds__(256) void chol_syrk(float* __restrict__ L, int jb) {
  int bi = jb + 1 + blockIdx.y;
  int bj = jb + 1 + blockIdx.x;
  if (bj > bi) return;
  extern __shared__ float lds[];
  float* Pa = lds;               // [128][129]
  float* Pb = lds + 128 * 129;   // [128][129]
  int tid = threadIdx.x;
  int b0 = jb * 128;
  for (int i = tid; i < 128 * 128; i += 256) {
    int r = i >> 7, c = i & 127;
    Pa[r * 129 + c] = L[(size_t)(bi * 128 + r) * MDIM + b0 + c];
    Pb[r * 129 + c] = L[(size_t)(bj * 128 + r) * MDIM + b0 + c];
  }
  __syncthreads();
  int ty = tid >> 4, tx = tid & 15;
  float acc[8][8] = {};
  for (int k = 0; k < 128; ++k) {
    float ra[8], rb[8];
#pragma unroll
    for (int a = 0; a < 8; ++a) ra[a] = Pa[(ty + a * 16) * 129 + k];
#pragma unroll
    for (int b = 0; b < 8; ++b) rb[b] = Pb[(tx + b * 16) * 129 + k];
#pragma unroll
    for (int a = 0; a < 8; ++a)
#pragma unroll
      for (int b = 0; b < 8; ++b) acc[a][b] += ra[a] * rb[b];
  }
#pragma unroll
  for (int a = 0; a < 8; ++a)
#pragma unroll
    for (int b = 0; b < 8; ++b) {
      size_t idx = (size_t)(bi * 128 + ty + a * 16) * MDIM + bj * 128 + tx + b * 16;
      L[idx] -= acc[a][b];
    }
}

// ---------------------------------------------------------------- transpose
__global__ __launch_bounds__(256) void transposeL(const float* __restrict__ L,
                                                  float* __restrict__ LT) {
  int idx = blockIdx.x * 256 + threadIdx.x;
  int i = idx >> 10, j = idx & 1023;
  LT[(size_t)j * MDIM + i] = L[(size_t)i * MDIM + j];
}

// ---------------------------- Kinv: one wave32 per column, fwd+bwd substitution
__global__ __launch_bounds__(256) void kinv_cols(const float* __restrict__ L,
                                                 const float* __restrict__ LT,
                                                 float* __restrict__ Kinv) {
  __shared__ float vbuf[8][MDIM];
  int lane = threadIdx.x & 31;
  int wave = threadIdx.x >> 5;
  int c = blockIdx.x * 8 + wave;
  volatile float* v = vbuf[wave];
  for (int i = lane; i < MDIM; i += 32) v[i] = 0.0f;
  if (lane == 0) v[c] = 1.0f;
  // forward: L v = e_c  (v[j]==0 for j<c)
  for (int j = c; j < MDIM; ++j) {
    float vj = v[j] / L[(size_t)j * MDIM + j];
    if (lane == 0) v[j] = vj;
    for (int i = j + 1 + lane; i < MDIM; i += 32)
      v[i] -= LT[(size_t)j * MDIM + i] * vj;
  }
  // backward: L^T w = v
  for (int j = MDIM - 1; j >= 0; --j) {
    float wj = v[j] / L[(size_t)j * MDIM + j];
    if (lane == 0) v[j] = wj;
    for (int i = lane; i < j; i += 32)
      v[i] -= L[(size_t)j * MDIM + i] * wj;
  }
  // Kinv symmetric -> write as row c (coalesced)
  for (int i = lane; i < MDIM; i += 32)
    Kinv[(size_t)c * MDIM + i] = v[i];
}

// -------------------------- beta = Kinv(mu_d - z), gamma = Kinv mu_d
__global__ __launch_bounds__(256) void small_solves(const float* __restrict__ Kinv,
                                                    const float* __restrict__ mu,
                                                    const float* __restrict__ z,
                                                    float* __restrict__ beta,
                                                    float* __restrict__ gamma) {
  int tid = threadIdx.x;
  int r = tid & 15;            // rhs index: 0..7 beta, 8..15 gamma
  int mLoc = tid >> 4;
  int m = blockIdx.x * 16 + mLoc;
  const float* Krow = Kinv + (size_t)m * MDIM;
  float acc = 0.0f;
  if (r < 8) {
    for (int k = 0; k < MDIM; ++k) acc += Krow[k] * (mu[r * MDIM + k] - z[k]);
    beta[r * MDIM + m] = acc;
  } else {
    int d = r - 8;
    for (int k = 0; k < MDIM; ++k) acc += Krow[k] * mu[d * MDIM + k];
    gamma[d * MDIM + m] = acc;
  }
}

// ---------------------------------------------------------------- main kernel
// Per WG: NT=32 columns. 8 waves; wave owns row-blocks rb = wave + 8*t and
// computes BOTH 16-column tiles from one A fragment (A reuse across N).
__global__ __launch_bounds__(256) void gp_main(const float* __restrict__ x,
                                               const float* __restrict__ z,
                                               const float* __restrict__ scales,
                                               const float* __restrict__ eps,
                                               const float* __restrict__ kinv,
                                               const float* __restrict__ beta,
                                               const float* __restrict__ ls_ptr,
                                               float* __restrict__ out) {
  extern __shared__ float lds[];
  float* sK = lds;                     // kzx^T : [NT][KPAD]
  float* sPart = lds + NT * KPAD;      // partials: [8 waves][2 cb][17][16]
  int tid = threadIdx.x;
  int n0 = blockIdx.x * NT;
  float invls = 1.0f / ls_ptr[0];

  // stage 0: build kzx^T tile in LDS (sK[n][m] = exp(-0.5*((z[m]-x[n0+n])/ls)^2))
  for (int i = tid; i < NT * MDIM; i += 256) {
    int n = i >> 10;
    int m = i & 1023;
    float d = (z[m] - x[n0 + n]) * invls;
    sK[n * KPAD + m] = __expf(-0.5f * d * d);
  }
  __syncthreads();

  // stage 1: alpha tile = Kinv @ kzx via fp32 WMMA, fused reductions
  int lane = tid & 31;
  int wave = tid >> 5;
  int l16 = lane & 15;
  int half = lane >> 4;               // lane half selects K offset / M offset

  float qacc[2] = {};
  float cacc[2][8] = {};
  float facc[2][8] = {};

  const float* Brow0 = sK + l16 * KPAD + half * 2;          // cb 0
  const float* Brow1 = sK + (16 + l16) * KPAD + half * 2;   // cb 1

  for (int t = 0; t < 8; ++t) {
    int rb = wave + 8 * t;                                  // row block 0..63
    const float* Arow = kinv + (size_t)(rb * 16 + l16) * MDIM + half * 2;
    v8f c0 = {};
    v8f c1 = {};
#pragma unroll 4
    for (int k0 = 0; k0 < MDIM; k0 += 4) {
      v2f a  = *(const v2f*)(Arow + k0);    // A: Kinv[row][k0+2h .. +1]
      v2f b0 = *(const v2f*)(Brow0 + k0);   // B: kzx[k0+2h .. +1][n]
      v2f b1 = *(const v2f*)(Brow1 + k0);
      c0 = __builtin_amdgcn_wmma_f32_16x16x4_f32(false, a, false, b0,
                                                 (short)0, c0, false, false);
      c1 = __builtin_amdgcn_wmma_f32_16x16x4_f32(false, a, false, b1,
                                                 (short)0, c1, false, false);
    }
    // epilogue: lane holds alpha[m][n], m = rb*16 + 8*half + v
    int mbase = rb * 16 + half * 8;
#pragma unroll
    for (int v = 0; v < 8; ++v) {
      int m = mbase + v;
      float alpha0 = c0[v];
      float alpha1 = c1[v];
      float kz0 = sK[l16 * KPAD + m];
      float kz1 = sK[(16 + l16) * KPAD + m];
      qacc[0] += alpha0 * kz0;
      qacc[1] += alpha1 * kz1;
      float a20 = alpha0 * alpha0;
      float a21 = alpha1 * alpha1;
#pragma unroll
      for (int d = 0; d < 8; ++d) {
        float s = scales[d * MDIM + m];
        float var = s * s;
        cacc[0][d] += var * a20;
        cacc[1][d] += var * a21;
        float b = beta[d * MDIM + m];
        facc[0][d] += b * kz0;
        facc[1][d] += b * kz1;
      }
    }
  }

  // combine lane halves (same n, complementary m halves)
#pragma unroll
  for (int cb = 0; cb < 2; ++cb) {
    qacc[cb] += __shfl_xor(qacc[cb], 16, 32);
#pragma unroll
    for (int d = 0; d < 8; ++d) {
      cacc[cb][d] += __shfl_xor(cacc[cb][d], 16, 32);
      facc[cb][d] += __shfl_xor(facc[cb][d], 16, 32);
    }
  }
  if (lane < 16) {
    int nl = lane;
#pragma unroll
    for (int cb = 0; cb < 2; ++cb) {
      int slot = wave * 2 + cb;
      sPart[(slot * 17 + 0) * 16 + nl] = qacc[cb];
#pragma unroll
      for (int d = 0; d < 8; ++d) {
        sPart[(slot * 17 + 1 + d) * 16 + nl] = cacc[cb][d];
        sPart[(slot * 17 + 9 + d) * 16 + nl] = facc[cb][d];
      }
    }
  }
  __syncthreads();

  // stage 2: deterministic combine of 8 wave-partials + output
  {
    int n = tid >> 3;
    int d = tid & 7;
    int cblk = n >> 4;
    int nl = n & 15;
    float q = 0.0f, corr = 0.0f, fm = 0.0f;
#pragma unroll
    for (int w = 0; w < 8; ++w) {
      int slot = w * 2 + cblk;
      q    += sPart[(slot * 17 + 0) * 16 + nl];
      corr += sPart[(slot * 17 + 1 + d) * 16 + nl];
      fm   += sPart[(slot * 17 + 9 + d) * 16 + nl];
    }
    float fmean = x[n0 + n] + fm;
    float fcov = 1.0f - q + corr;
    fcov = fcov > 0.0f ? fcov : 0.0f;
    out[(size_t)(n0 + n) * DOUT + d] =
        fmean + sqrtf(fcov) * eps[(size_t)d * NTOT + n0 + n];
  }
}

// ---------------------------------------------------------------- KL scalar
__global__ __launch_bounds__(1024) void kl_kernel(const float* __restrict__ L,
                                                  const float* __restrict__ Kinv,
                                                  const float* __restrict__ mu,
                                                  const float* __restrict__ scales,
                                                  const float* __restrict__ gamma,
                                                  float* __restrict__ out_kl) {
  __shared__ float red[1024];
  int m = threadIdx.x;
  float var_sum = 0.0f, iq = 0.0f, ld0 = 0.0f;
#pragma unroll
  for (int d = 0; d < 8; ++d) {
    float s = scales[d * MDIM + m];
    var_sum += s * s;
    iq += mu[d * MDIM + m] * gamma[d * MDIM + m];
    ld0 += __logf(s * s);
  }
  float tr = var_sum * Kinv[(size_t)m * MDIM + m];
  float l1 = __logf(L[(size_t)m * MDIM + m]);
  red[m] = 8.0f * tr + iq + 16.0f * l1 - ld0;   // D*trace + invquad + D*2*logL - logdet0
  __syncthreads();
  for (int s2 = 512; s2 > 0; s2 >>= 1) {
    if (m < s2) red[m] += red[m + s2];
    __syncthreads();
  }
  if (m == 0) out_kl[0] = 0.5f * (red[0] - (float)(DOUT * MDIM));
}

// ---------------------------------------------------------------- launcher
extern "C" void kernel_launch(void* const* d_in, const int* in_sizes, int n_in,
                              void* d_out, int out_size, void* d_ws, size_t ws_size,
                              hipStream_t stream) {
  const float* x      = (const float*)d_in[0];
  const float* z      = (const float*)d_in[1];
  const float* mu     = (const float*)d_in[2];
  const float* scales = (const float*)d_in[3];
  const float* ls     = (const float*)d_in[4];
  const float* eps    = (const float*)d_in[5];
  float* out = (float*)d_out;

  char* ws = (char*)d_ws;
  float* L     = (float*)(ws);                            // 4 MB
  float* LT    = (float*)(ws + ((size_t)4 << 20));        // 4 MB
  float* Kinv  = (float*)(ws + ((size_t)8 << 20));        // 4 MB
  float* beta  = (float*)(ws + ((size_t)12 << 20));       // 32 KB
  float* gamma = beta + DOUT * MDIM;                      // 32 KB

  build_kzz<<<MDIM, 256, 0, stream>>>(z, ls, L);

  for (int jb = 0; jb < 8; ++jb) {
    chol_diag<<<1, 256, 128 * 129 * 4, stream>>>(L, jb);
    int rem = 7 - jb;
    if (rem > 0) {
      chol_trsm<<<rem, 128, 2 * 128 * 129 * 4, stream>>>(L, jb);
      chol_syrk<<<dim3(rem, rem), 256, 2 * 128 * 129 * 4, stream>>>(L, jb);
    }
  }

  transposeL<<<(MDIM * MDIM) / 256, 256, 0, stream>>>(L, LT);
  kinv_cols<<<MDIM / 8, 256, 0, stream>>>(L, LT, Kinv);
  small_solves<<<MDIM / 16, 256, 0, stream>>>(Kinv, mu, z, beta, gamma);

  size_t mainLds = (size_t)(NT * KPAD + 16 * 17 * 16) * 4;  // 148,992 B
  gp_main<<<NTOT / NT, 256, mainLds, stream>>>(x, z, scales, eps, Kinv, beta, ls, out);

  kl_kernel<<<1, 1024, 0, stream>>>(L, Kinv, mu, scales, gamma,
                                    out + (size_t)NTOT * DOUT);
}